// OptimizedSpatialCorrelationSampler_44255343018945
// MI455X (gfx1250) — compile-verified
//
#include <hip/hip_runtime.h>

// Problem dims (fixed by the reference)
#define BN    16
#define CN    512
#define HN    38
#define WN    68
#define PADN  4
#define KN    9
#define PHN   (HN + 2 * PADN)   // 46 padded rows
#define PLANE (HN * WN)         // 2584 elements per (b,c) plane
#define NROWS (PHN * WN)        // 3128 row-stat entries
#define NTH   256
#define NCHUNK (PLANE / 4)      // 646 float4 chunks (exact: 2584 % 4 == 0)

#if __has_builtin(__builtin_amdgcn_global_load_async_to_lds_b128)
#define HAVE_ASYNC_LDS 1
#else
#define HAVE_ASYNC_LDS 0
#endif

// Match the builtin's parameter types exactly:
//   (v4i AS(1)*, v4i AS(3)*, imm int, imm int)
typedef int v4i __attribute__((vector_size(4 * sizeof(int))));
typedef __attribute__((address_space(1))) v4i gv4i_t;  // global
typedef __attribute__((address_space(3))) v4i lv4i_t;  // LDS

// Async DMA of 16 bytes global -> LDS (CDNA5 path); sync fallback otherwise.
__device__ __forceinline__ void async_b128(const float* g, float* l) {
#if HAVE_ASYNC_LDS
  // Generic global pointers are bitwise-identical to AS(1); generic LDS
  // pointers carry the LDS byte offset in their low 32 bits, so integer
  // round-trips produce valid AS pointers without addrspacecast syntax.
  __builtin_amdgcn_global_load_async_to_lds_b128(
      (gv4i_t*)(unsigned long long)g,
      (lv4i_t*)(unsigned int)(unsigned long long)l,
      /*offset=*/0, /*cpol=*/0);
#else
  *(float4*)l = *(const float4*)g;
#endif
}

__device__ __forceinline__ void wait_async_done() {
#if HAVE_ASYNC_LDS
#if __has_builtin(__builtin_amdgcn_s_wait_asynccnt)
  __builtin_amdgcn_s_wait_asynccnt(0);
#else
  asm volatile("s_wait_asynccnt 0" ::: "memory");
#endif
#endif
}

__global__ __launch_bounds__(NTH) void
corr_max_kernel(const float* __restrict__ f1,
                const float* __restrict__ f2,
                float* __restrict__ out) {
  __shared__ __align__(16) float s_f2[PLANE];   // feat2 plane copy
  __shared__ float s_rmax[NROWS];               // row-wise 9-tap max (padded rows)
  __shared__ float s_rmin[NROWS];               // row-wise 9-tap min

  const int tid = threadIdx.x;
  const long long base = (long long)blockIdx.x * PLANE;  // (b*C + c) plane

  // ---- Stage feat2 plane into LDS via async global->LDS DMA ----
  {
    const float* g = f2 + base;
    for (int i = tid; i < NCHUNK; i += NTH)
      async_b128(g + i * 4, s_f2 + i * 4);
    wait_async_done();
  }
  __syncthreads();

  // ---- Row pass: 9-wide max/min per padded row (zero padding included) ----
  for (int i = tid; i < NROWS; i += NTH) {
    const int py = i / WN;        // padded row index 0..45
    const int x  = i - py * WN;   // output column 0..67
    const int sy = py - PADN;     // source row in feat2
    float mx = 0.0f, mn = 0.0f;   // fully-padded rows are all zeros
    if (sy >= 0 && sy < HN) {
      mx = -__builtin_inff();
      mn =  __builtin_inff();
      const float* row = &s_f2[sy * WN];
#pragma unroll
      for (int dx = 0; dx < KN; ++dx) {
        const int xx = x - PADN + dx;                       // padded col - PAD
        const float v = (xx >= 0 && xx < WN) ? row[xx] : 0.0f;
        mx = fmaxf(mx, v);
        mn = fminf(mn, v);
      }
    }
    s_rmax[i] = mx;
    s_rmin[i] = mn;
  }
  __syncthreads();

  // ---- Column pass + combine: out = max(a*winmax, a*winmin) ----
  {
    const float4* g1 = (const float4*)(f1 + base);
    float4* go = (float4*)(out + base);
    for (int i = tid; i < NCHUNK; i += NTH) {
      const int e = i * 4;            // WN % 4 == 0 -> all 4 lanes in one row
      const int h = e / WN;
      const int w = e - h * WN;
      float cmx[4], cmn[4];
#pragma unroll
      for (int j = 0; j < 4; ++j) {
        cmx[j] = -__builtin_inff();
        cmn[j] =  __builtin_inff();
      }
#pragma unroll
      for (int dy = 0; dy < KN; ++dy) {
        const int r = (h + dy) * WN + w;   // padded row h+dy in 0..45
#pragma unroll
        for (int j = 0; j < 4; ++j) {
          cmx[j] = fmaxf(cmx[j], s_rmax[r + j]);
          cmn[j] = fminf(cmn[j], s_rmin[r + j]);
        }
      }
      const float4 a = g1[i];
      float4 o;
      o.x = fmaxf(a.x * cmx[0], a.x * cmn[0]);
      o.y = fmaxf(a.y * cmx[1], a.y * cmn[1]);
      o.z = fmaxf(a.z * cmx[2], a.z * cmn[2]);
      o.w = fmaxf(a.w * cmx[3], a.w * cmn[3]);
      go[i] = o;
    }
  }
}

extern "C" void kernel_launch(void* const* d_in, const int* in_sizes, int n_in,
                              void* d_out, int out_size, void* d_ws, size_t ws_size,
                              hipStream_t stream) {
  (void)in_sizes; (void)n_in; (void)out_size; (void)d_ws; (void)ws_size;
  const float* f1 = (const float*)d_in[0];   // feat1
  const float* f2 = (const float*)d_in[1];   // feat2
  float* out = (float*)d_out;                // [16,512,38,68] f32
  corr_max_kernel<<<dim3(BN * CN), dim3(NTH), 0, stream>>>(f1, f2, out);
}